// APRMaxPool_5257039970544
// MI455X (gfx1250) — compile-verified
//
#include <hip/hip_runtime.h>
#include <cfloat>
#include <cstdint>

// APR tree-level max-pool (segment max with sorted segment ids).
//
// Memory-bound: 256 MiB streaming input (NT b128 loads, read-once > 192MB L2),
// 8 MiB shared ids (staged in LDS via CDNA5 async global->LDS copy, reused by
// all 64 channels), 32 MiB output resolved with native non-returning
// global_atomic_max_num_f32 at device scope (RMW in the L2 atomic units).
// HBM roofline ~13 us @ 23.3 TB/s.

#define TPB  256          // threads per block (8 wave32 waves)
#define VEC  4            // elements per thread per channel (float4)
#define TILE (TPB * VEC)  // 1024 input particles per block

typedef float v4f __attribute__((ext_vector_type(4)));

// Native CDNA5 float atomic max, non-returning, device scope.
// base must be uniform (lands in an SGPR pair); byteoff/val are per-lane.
__device__ __forceinline__ void atomic_max_f32(const float* base,
                                               unsigned byteoff, float val) {
    asm volatile("global_atomic_max_num_f32 %0, %1, %2 scope:SCOPE_DEV"
                 :: "v"(byteoff), "v"(val),
                    "s"((unsigned long long)(uintptr_t)base)
                 : "memory");
}

__global__ __launch_bounds__(TPB)
void apr_init(float* __restrict__ out, long long n) {
    long long i = ((long long)blockIdx.x * TPB + threadIdx.x) * 4;
    if (i + 3 < n) {
        v4f v = {-FLT_MAX, -FLT_MAX, -FLT_MAX, -FLT_MAX};
        *(v4f*)(out + i) = v;
    } else {
        for (long long k = i; k < n; ++k) out[k] = -FLT_MAX;
    }
}

__global__ __launch_bounds__(TPB)
void apr_pool(const float* __restrict__ x,          // [nch, n_in]
              const long long* __restrict__ pid,    // [n_in], sorted
              float* __restrict__ out,              // [nch, n_out]
              int nch, long long n_in, long long n_out) {
    __shared__ long long sid[TILE];   // staged parent ids (raw int64)

    const int t = threadIdx.x;
    const long long n0 = (long long)blockIdx.x * TILE;

    // ---- Stage this tile's parent ids into LDS via the CDNA5 async path.
    // GVS mode: 64-bit SGPR base + per-lane 32-bit VGPR byte offset.
    {
        unsigned ldsbase = (unsigned)(uintptr_t)(void*)&sid[0];
        unsigned long long gbase = (unsigned long long)(uintptr_t)(pid + n0);
#pragma unroll
        for (int k = 0; k < TILE / TPB; ++k) {
            unsigned e    = (unsigned)t + (unsigned)k * TPB;
            unsigned lds  = ldsbase + e * 8u;
            unsigned voff = e * 8u;
            asm volatile("global_load_async_to_lds_b64 %0, %1, %2"
                         :: "v"(lds), "v"(voff), "s"(gbase)
                         : "memory");
        }
        asm volatile("s_wait_asynccnt 0x0" ::: "memory");
    }
    __syncthreads();

    // ---- Each thread owns 4 consecutive particles; ids from LDS.
    const int base = t * VEC;
    const unsigned off0 = (unsigned)(int)sid[base + 0] * 4u;
    const unsigned off1 = (unsigned)(int)sid[base + 1] * 4u;
    const unsigned off2 = (unsigned)(int)sid[base + 2] * 4u;
    const unsigned off3 = (unsigned)(int)sid[base + 3] * 4u;

    const float* px = x + n0 + base;

    // Loop over all B*C channels (structure shared across channels).
    // NT loads: input is streamed exactly once; keep it out of L2 so the
    // output atomics stay resident there instead.
#pragma unroll 4
    for (int bc = 0; bc < nch; ++bc) {
        v4f v = __builtin_nontemporal_load((const v4f*)(px + (long long)bc * n_in));
        const float* o = out + (long long)bc * n_out;

        // Run-compress the 4 sorted ids -> ~1-2 atomics instead of 4.
        float m = v.x;
        unsigned cur = off0;
        if (off1 != cur) { atomic_max_f32(o, cur, m); cur = off1; m = v.y; }
        else             { m = fmaxf(m, v.y); }
        if (off2 != cur) { atomic_max_f32(o, cur, m); cur = off2; m = v.z; }
        else             { m = fmaxf(m, v.z); }
        if (off3 != cur) { atomic_max_f32(o, cur, m); cur = off3; m = v.w; }
        else             { m = fmaxf(m, v.w); }
        atomic_max_f32(o, cur, m);
    }
}

// Scalar fallback for a partial tile at the end (not launched when
// n_in is a multiple of TILE, as with N_IN = 1M).
__global__ __launch_bounds__(TPB)
void apr_tail(const float* __restrict__ x, const long long* __restrict__ pid,
              float* __restrict__ out, int nch, long long n_in, long long n_out,
              long long start) {
    long long i = start + (long long)blockIdx.x * TPB + threadIdx.x;
    if (i >= n_in) return;
    unsigned off = (unsigned)(int)pid[i] * 4u;
    for (int bc = 0; bc < nch; ++bc) {
        float v = x[(long long)bc * n_in + i];
        atomic_max_f32(out + (long long)bc * n_out, off, v);
    }
}

extern "C" void kernel_launch(void* const* d_in, const int* in_sizes, int n_in_args,
                              void* d_out, int out_size, void* d_ws, size_t ws_size,
                              hipStream_t stream) {
    (void)d_ws; (void)ws_size; (void)n_in_args;

    const float*     x   = (const float*)d_in[0];      // [B, C, N_IN] f32
    const long long* pid = (const long long*)d_in[1];  // [N_IN] i64, sorted
    float*           out = (float*)d_out;              // [B, C, N_OUT] f32

    const long long n_in_total = (long long)in_sizes[0];   // B*C*N_IN
    const long long n_in       = (long long)in_sizes[1];   // N_IN
    const int       nch        = (int)(n_in_total / n_in); // B*C = 64
    const long long n_out      = (long long)out_size / nch;// N_OUT

    // 1) Fill output with -FLT_MAX (empty-segment fill value).
    {
        long long total = (long long)out_size;
        int blocks = (int)((total / 4 + TPB - 1) / TPB);
        apr_init<<<blocks, TPB, 0, stream>>>(out, total);
    }

    // 2) Segmented atomic max over full tiles.
    {
        long long tiles = n_in / TILE;
        if (tiles > 0)
            apr_pool<<<(int)tiles, TPB, 0, stream>>>(x, pid, out, nch, n_in, n_out);

        long long rem = n_in - tiles * TILE;
        if (rem > 0) {
            int blocks = (int)((rem + TPB - 1) / TPB);
            apr_tail<<<blocks, TPB, 0, stream>>>(x, pid, out, nch, n_in, n_out,
                                                 tiles * TILE);
        }
    }
}